// SelfAttentionCNN_90675349553771
// MI455X (gfx1250) — compile-verified
//
#include <hip/hip_runtime.h>
#include <math.h>

// ---------------------------------------------------------------------------
// Problem constants (from reference): B=32, H=128, W=128, C=64, N=8
// ---------------------------------------------------------------------------
#define NPIX   (32 * 128 * 128)   // 524288 pixels (B*H*W)
#define CCH    64                 // channels
#define NHEADS 8                  // attention heads
#define TOTAL_ELEMS ((size_t)NPIX * CCH)

typedef __attribute__((ext_vector_type(2))) float v2f;
typedef __attribute__((ext_vector_type(8))) float v8f;

// Gram pass configuration
#define K1_WGS          256
#define K1_WAVES_PER_WG 4
#define K1_TOTAL_WAVES  (K1_WGS * K1_WAVES_PER_WG)   // 1024 waves
#define PIX_PER_WAVE    (NPIX / K1_TOTAL_WAVES)      // 512 pixels per wave
#define K1_STEPS        (PIX_PER_WAVE / 4)           // 128 K=4 steps

// ---------------------------------------------------------------------------
// Kernel 1: partial Gram matrices  M = X^T X  via V_WMMA_F32_16X16X4_F32.
// X is [NPIX, 64] row-major.  Tile: D(16x16) += A(16x4) * B(4x16) where
// A = X^T chunk (M=channel, K=pixel) and B = X chunk (K=pixel, N=channel).
// Per ISA layouts, A-frag and B-frag for the same channel group are the SAME
// per-lane data:  lane(l): value[v] = X[p0 + 2*(l>>4) + v, cgrp + (l&15)].
// So each 4-pixel step loads 4 fragments (one per 16-channel group) and
// issues 16 WMMAs covering the full 64x64 Gram tile.
// ---------------------------------------------------------------------------
__global__ __launch_bounds__(128) void gram_wmma_kernel(
    const float* __restrict__ x, float* __restrict__ partials) {
  __shared__ float red[K1_WAVES_PER_WG * CCH * CCH];   // 64 KB

  const int lane     = threadIdx.x & 31;
  const int waveInWg = threadIdx.x >> 5;
  const int wave     = blockIdx.x * K1_WAVES_PER_WG + waveInWg;
  const int laneM    = lane & 15;   // channel within group / N column
  const int laneG    = lane >> 4;   // pixel-pair selector (K base 0 or 2)

  const v8f vzero = {};
  v8f acc[4][4];
#pragma unroll
  for (int i = 0; i < 4; ++i)
#pragma unroll
    for (int j = 0; j < 4; ++j) acc[i][j] = vzero;

  const size_t p0 = (size_t)wave * PIX_PER_WAVE;
  // lane base: pixel (p0 + 2*laneG), channel laneM
  const float* base = x + (p0 + (size_t)(laneG * 2)) * CCH + laneM;

  for (int s = 0; s < K1_STEPS; ++s) {
    const float* bp = base + (size_t)s * (4 * CCH);
    v2f f[4];
#pragma unroll
    for (int cg = 0; cg < 4; ++cg) {
      f[cg].x = bp[cg * 16];          // K = 2*laneG + 0
      f[cg].y = bp[cg * 16 + CCH];    // K = 2*laneG + 1
    }
#pragma unroll
    for (int i = 0; i < 4; ++i)
#pragma unroll
      for (int j = 0; j < 4; ++j)
        acc[i][j] = __builtin_amdgcn_wmma_f32_16x16x4_f32(
            false, f[i], false, f[j], (short)0, acc[i][j], false, false);
  }

  // Spill accumulators to LDS.  C/D layout: elem r of lane l -> row = r +
  // 8*(l>>4), col = l&15 within the 16x16 tile.
  float* myred = red + waveInWg * (CCH * CCH);
#pragma unroll
  for (int i = 0; i < 4; ++i)
#pragma unroll
    for (int j = 0; j < 4; ++j)
#pragma unroll
      for (int r = 0; r < 8; ++r) {
        const int row = i * 16 + r + laneG * 8;
        const int col = j * 16 + laneM;
        myred[row * CCH + col] = acc[i][j][r];
      }
  __syncthreads();

  // 4 waves -> 1 partial Gram per workgroup
  float* outp = partials + (size_t)blockIdx.x * (CCH * CCH);
  for (int e = threadIdx.x; e < CCH * CCH; e += 128) {
    outp[e] = red[e] + red[CCH * CCH + e] + red[2 * CCH * CCH + e] +
              red[3 * CCH * CCH + e];
  }
}

// ---------------------------------------------------------------------------
// Kernel 2: reduce 256 partial Gram matrices -> M[64*64]
// ---------------------------------------------------------------------------
__global__ __launch_bounds__(256) void gram_reduce_kernel(
    const float* __restrict__ partials, float* __restrict__ M) {
  const int e = blockIdx.x * 256 + threadIdx.x;  // 0..4095
  float s = 0.f;
  for (int w = 0; w < K1_WGS; ++w) s += partials[(size_t)w * (CCH * CCH) + e];
  M[e] = s;
}

// ---------------------------------------------------------------------------
// Kernel 3: ST = Wg^T * M * Wf, row softmax -> Beta, then
//           w_eff[c] = Gamma * sum_j (sum_i Wv[i]*Beta[i,j]) * Wh[c,j]
// One block of 512 threads; all contractions are tiny.
// ---------------------------------------------------------------------------
__global__ __launch_bounds__(512) void attn_small_kernel(
    const float* __restrict__ M, const float* __restrict__ Wf,
    const float* __restrict__ Wg, const float* __restrict__ Wh,
    const float* __restrict__ Wv, const float* __restrict__ Gamma,
    float* __restrict__ w_eff) {
  __shared__ float sWf[CCH * NHEADS];     // 512
  __shared__ float sT[CCH * NHEADS];      // T[c1,j] = sum_c2 M[c1,c2]Wf[c2,j]
  __shared__ float sST[NHEADS * NHEADS];
  __shared__ float sBeta[NHEADS * NHEADS];
  __shared__ float sU[NHEADS];

  const int t = threadIdx.x;
  sWf[t] = Wf[t];  // blockDim == 512 == CCH*NHEADS
  __syncthreads();

  {  // T[c1,j]
    const int c1 = t >> 3, j = t & 7;
    float s = 0.f;
    for (int c2 = 0; c2 < CCH; ++c2)
      s += M[c1 * CCH + c2] * sWf[c2 * NHEADS + j];
    sT[c1 * NHEADS + j] = s;
  }
  __syncthreads();

  if (t < NHEADS * NHEADS) {  // ST[i,j] = sum_c1 Wg[c1,i] * T[c1,j]
    const int i = t >> 3, j = t & 7;
    float s = 0.f;
    for (int c1 = 0; c1 < CCH; ++c1)
      s += Wg[c1 * NHEADS + i] * sT[c1 * NHEADS + j];
    sST[i * NHEADS + j] = s;
  }
  __syncthreads();

  if (t < NHEADS) {  // row-wise softmax
    float m = sST[t * NHEADS];
    for (int j = 1; j < NHEADS; ++j) m = fmaxf(m, sST[t * NHEADS + j]);
    float e[NHEADS], sum = 0.f;
    for (int j = 0; j < NHEADS; ++j) {
      e[j] = expf(sST[t * NHEADS + j] - m);
      sum += e[j];
    }
    const float inv = 1.f / sum;
    for (int j = 0; j < NHEADS; ++j) sBeta[t * NHEADS + j] = e[j] * inv;
  }
  __syncthreads();

  if (t < NHEADS) {  // u[j] = sum_i Wv[i] * Beta[i,j]
    float s = 0.f;
    for (int i = 0; i < NHEADS; ++i) s += Wv[i] * sBeta[i * NHEADS + t];
    sU[t] = s;
  }
  __syncthreads();

  if (t < CCH) {  // w_eff[c] = Gamma * sum_j u[j] * Wh[c,j]
    float s = 0.f;
    for (int j = 0; j < NHEADS; ++j) s += sU[j] * Wh[t * NHEADS + j];
    w_eff[t] = s * Gamma[0];
  }
}

// ---------------------------------------------------------------------------
// Kernel 4: V[p] = dot(x[p, 0:64], w_eff)   (one thread per pixel)
// ---------------------------------------------------------------------------
__global__ __launch_bounds__(256) void v_kernel(
    const float* __restrict__ x, const float* __restrict__ w_eff,
    float* __restrict__ V) {
  __shared__ float w[CCH];
  if (threadIdx.x < CCH) w[threadIdx.x] = w_eff[threadIdx.x];
  __syncthreads();

  const size_t p = (size_t)blockIdx.x * 256 + threadIdx.x;
  const float4* xp = (const float4*)(x + p * CCH);
  const float4* wp = (const float4*)w;
  float acc = 0.f;
#pragma unroll
  for (int i = 0; i < CCH / 4; ++i) {
    const float4 xv = xp[i];
    const float4 wv = wp[i];
    acc = fmaf(xv.x, wv.x, acc);
    acc = fmaf(xv.y, wv.y, acc);
    acc = fmaf(xv.z, wv.z, acc);
    acc = fmaf(xv.w, wv.w, acc);
  }
  V[p] = acc;
}

// ---------------------------------------------------------------------------
// Kernel 5: out.flat[k] = x.flat[k] + V[k mod NPIX]   (float4; NPIX % 4 == 0)
// ---------------------------------------------------------------------------
__global__ __launch_bounds__(256) void out_kernel(
    const float* __restrict__ x, const float* __restrict__ V,
    float* __restrict__ out) {
  const size_t t = (size_t)blockIdx.x * 256 + threadIdx.x;  // float4 index
  const float4 xv = ((const float4*)x)[t];
  const float4 vv = ((const float4*)V)[t & (NPIX / 4 - 1)];  // NPIX/4 is pow2
  float4 o;
  o.x = xv.x + vv.x;
  o.y = xv.y + vv.y;
  o.z = xv.z + vv.z;
  o.w = xv.w + vv.w;
  ((float4*)out)[t] = o;
}

// ---------------------------------------------------------------------------
// Launcher.  Inputs: x[33554432], Wf[512], Wg[512], Wh[512], Wv[8], Gamma[1].
// Workspace layout (floats): V[NPIX] | M[4096] | w_eff[64] | partials[256*4096]
// Total ~6.3 MB.
// ---------------------------------------------------------------------------
extern "C" void kernel_launch(void* const* d_in, const int* in_sizes, int n_in,
                              void* d_out, int out_size, void* d_ws,
                              size_t ws_size, hipStream_t stream) {
  const float* x     = (const float*)d_in[0];
  const float* Wf    = (const float*)d_in[1];
  const float* Wg    = (const float*)d_in[2];
  const float* Wh    = (const float*)d_in[3];
  const float* Wv    = (const float*)d_in[4];
  const float* Gamma = (const float*)d_in[5];
  float* out = (float*)d_out;

  float* ws       = (float*)d_ws;
  float* V        = ws;                       // NPIX
  float* M        = V + NPIX;                 // 4096
  float* w_eff    = M + CCH * CCH;            // 64
  float* partials = w_eff + CCH;              // K1_WGS * 4096

  gram_wmma_kernel<<<K1_WGS, 128, 0, stream>>>(x, partials);
  gram_reduce_kernel<<<(CCH * CCH) / 256, 256, 0, stream>>>(partials, M);
  attn_small_kernel<<<1, 512, 0, stream>>>(M, Wf, Wg, Wh, Wv, Gamma, w_eff);
  v_kernel<<<NPIX / 256, 256, 0, stream>>>(x, w_eff, V);
  out_kernel<<<(int)(TOTAL_ELEMS / 4 / 256), 256, 0, stream>>>(x, V, out);
}